// InteractionBlock_79654463472005
// MI455X (gfx1250) — compile-verified
//
#include <hip/hip_runtime.h>
#include <math.h>

// ---------------------------------------------------------------------------
// MI455X (gfx1250) implementation of the e3nn-style interaction block.
// All matrix math uses V_WMMA_F32_16X16X4_F32 (wave32, fp32 in/out).
// ---------------------------------------------------------------------------

#define NN 16384      // nodes
#define EE 262144     // edges
#define MUL 64
#define RDIM 8
#define NEURONS 64
#define NSPEC 4

typedef float v2f __attribute__((ext_vector_type(2)));
typedef float v8f __attribute__((ext_vector_type(8)));

__device__ __forceinline__ v8f wmma4(v2f a, v2f b, v8f c) {
  // D(16x16,f32) = A(16x4,f32) * B(4x16,f32) + C
  return __builtin_amdgcn_wmma_f32_16x16x4_f32(
      /*neg_a=*/false, a, /*neg_b=*/false, b,
      /*c_mod=*/(short)0, c, /*reuse_a=*/false, /*reuse_b=*/false);
}

__device__ __forceinline__ void wave_lds_sync() {
  // Per-wave LDS buffers only: DS ops are in-order per wave; we only need the
  // outstanding ds stores to land before cross-lane ds loads.
  asm volatile("s_wait_dscnt 0" ::: "memory");
}

__device__ __forceinline__ float silu(float x) {
  return x / (1.0f + __expf(-x));
}

// ---------------------------------------------------------------------------
// Kernel 1: linear_up.  h0 = x0 @ Wup0 / 8 ;  h1[:,:,c] = x1[:,:,c] @ Wup1 / 8
// One wave per 16-node tile.  1024 tiles total.
// ---------------------------------------------------------------------------
__global__ __launch_bounds__(256) void k_linear_up(
    const float* __restrict__ x0, const float* __restrict__ x1,
    const float* __restrict__ Wup0, const float* __restrict__ Wup1,
    float* __restrict__ h0, float* __restrict__ h1) {
  const int lane = threadIdx.x & 31;
  const int wave = threadIdx.x >> 5;
  const int half = lane >> 4;     // K/row-half selector
  const int lcol = lane & 15;     // A row / B-D column within tile
  const int nwaves = (blockDim.x >> 5) * gridDim.x;
  const int gwave = blockIdx.x * (blockDim.x >> 5) + wave;
  const float inv = 0.125f;       // 1/sqrt(MUL)

  for (int tile = gwave; tile < NN / 16; tile += nwaves) {
    const int n0 = tile * 16;
    // ---- h0 ----
    for (int ct = 0; ct < 4; ++ct) {
      v8f acc = {};
      for (int k0 = 0; k0 < MUL; k0 += 4) {
        const int kk = k0 + 2 * half;
        v2f a; a.x = x0[(n0 + lcol) * MUL + kk];
               a.y = x0[(n0 + lcol) * MUL + kk + 1];
        v2f b; b.x = Wup0[kk * MUL + ct * 16 + lcol];
               b.y = Wup0[(kk + 1) * MUL + ct * 16 + lcol];
        acc = wmma4(a, b, acc);
      }
#pragma unroll
      for (int i = 0; i < 8; ++i) {
        const int row = n0 + i + 8 * half;
        h0[row * MUL + ct * 16 + lcol] = acc[i] * inv;
      }
    }
    // ---- h1 (3 components) ----
    for (int c = 0; c < 3; ++c) {
      for (int ct = 0; ct < 4; ++ct) {
        v8f acc = {};
        for (int k0 = 0; k0 < MUL; k0 += 4) {
          const int kk = k0 + 2 * half;
          v2f a; a.x = x1[((n0 + lcol) * MUL + kk) * 3 + c];
                 a.y = x1[((n0 + lcol) * MUL + kk + 1) * 3 + c];
          v2f b; b.x = Wup1[kk * MUL + ct * 16 + lcol];
                 b.y = Wup1[(kk + 1) * MUL + ct * 16 + lcol];
          acc = wmma4(a, b, acc);
        }
#pragma unroll
        for (int i = 0; i < 8; ++i) {
          const int row = n0 + i + 8 * half;
          h1[(row * MUL + ct * 16 + lcol) * 3 + c] = acc[i] * inv;
        }
      }
    }
  }
}

// ---------------------------------------------------------------------------
// Kernel 2: radial MLP (WMMA) + CG tensor product + atomic scatter.
// One wave handles 16-edge subtiles; per-wave LDS for the C->A layout change.
// ---------------------------------------------------------------------------
#define K2_WAVES 8
__global__ __launch_bounds__(K2_WAVES * 32) void k_edge(
    const float* __restrict__ re, const float* __restrict__ Rw1,
    const float* __restrict__ Rw2,
    const float* __restrict__ h0, const float* __restrict__ h1,
    const float* __restrict__ y0, const float* __restrict__ y1,
    const int* __restrict__ senders, const int* __restrict__ receivers,
    float* __restrict__ a0, float* __restrict__ a1) {
  __shared__ float s_hid[K2_WAVES][16 * NEURONS];  // 4 KB / wave
  __shared__ float s_wj[K2_WAVES][16 * MUL];       // 4 KB / wave
  const int lane = threadIdx.x & 31;
  const int wave = threadIdx.x >> 5;
  const int half = lane >> 4;
  const int lcol = lane & 15;
  const int nwaves = K2_WAVES * gridDim.x;
  const int gwave = blockIdx.x * K2_WAVES + wave;
  const float rs1 = 0.35355339059327373f;  // 1/sqrt(RDIM)
  const float rs2 = 0.125f;                // 1/sqrt(NEURONS)
  const float invsq3 = 0.5773502691896258f;
  const float invden = 0.25f;              // 1/sqrt(AVG_NEIGH)

  for (int st = gwave; st < EE / 16; st += nwaves) {
    const int e0 = st * 16;
    // ---- hid = silu((re @ Rw1)/sqrt(8))  [16 x 64] -> LDS ----
    for (int ct = 0; ct < 4; ++ct) {
      v8f acc = {};
      for (int k0 = 0; k0 < RDIM; k0 += 4) {
        const int kk = k0 + 2 * half;
        v2f a; a.x = re[(e0 + lcol) * RDIM + kk];
               a.y = re[(e0 + lcol) * RDIM + kk + 1];
        v2f b; b.x = Rw1[kk * NEURONS + ct * 16 + lcol];
               b.y = Rw1[(kk + 1) * NEURONS + ct * 16 + lcol];
        acc = wmma4(a, b, acc);
      }
#pragma unroll
      for (int i = 0; i < 8; ++i)
        s_hid[wave][(i + 8 * half) * NEURONS + ct * 16 + lcol] =
            silu(acc[i] * rs1);
    }
    wave_lds_sync();

    // ---- per weight-group j: w_j = (hid @ Rw2[:, j*64:+64])/8, then TP ----
    for (int j = 0; j < 4; ++j) {
      for (int ct = 0; ct < 4; ++ct) {
        v8f acc = {};
        for (int k0 = 0; k0 < NEURONS; k0 += 4) {
          const int kk = k0 + 2 * half;
          v2f a; a.x = s_hid[wave][lcol * NEURONS + kk];
                 a.y = s_hid[wave][lcol * NEURONS + kk + 1];
          v2f b; b.x = Rw2[kk * 256 + j * 64 + ct * 16 + lcol];
                 b.y = Rw2[(kk + 1) * 256 + j * 64 + ct * 16 + lcol];
          acc = wmma4(a, b, acc);
        }
#pragma unroll
        for (int i = 0; i < 8; ++i)
          s_wj[wave][(i + 8 * half) * MUL + ct * 16 + lcol] = acc[i] * rs2;
      }
      wave_lds_sync();

      // ---- tensor-product term j + scatter (L2-resident atomics) ----
      for (int e = 0; e < 16; ++e) {
        const int eg = e0 + e;
        const int snd = senders[eg];
        const int rcv = receivers[eg];
        const float ey0 = y0[eg];
        const float e1x = y1[eg * 3 + 0];
        const float e1y = y1[eg * 3 + 1];
        const float e1z = y1[eg * 3 + 2];
#pragma unroll
        for (int mh = 0; mh < 2; ++mh) {
          const int m = lane + 32 * mh;
          const float wv = s_wj[wave][e * MUL + m] * invden;
          if (j == 0) {                       // 0e x 0e -> 0e
            const float s0v = h0[snd * MUL + m];
            atomicAdd(&a0[rcv * 128 + m], wv * s0v * ey0);
          } else if (j == 1) {                // 1o . 1o -> 0e
            const float* s1p = &h1[(snd * MUL + m) * 3];
            const float dot = s1p[0] * e1x + s1p[1] * e1y + s1p[2] * e1z;
            atomicAdd(&a0[rcv * 128 + 64 + m], wv * dot * invsq3);
          } else if (j == 2) {                // 0e x 1o -> 1o
            const float base = wv * h0[snd * MUL + m];
            atomicAdd(&a1[(rcv * 128 + m) * 3 + 0], base * e1x);
            atomicAdd(&a1[(rcv * 128 + m) * 3 + 1], base * e1y);
            atomicAdd(&a1[(rcv * 128 + m) * 3 + 2], base * e1z);
          } else {                            // 1o x 0e -> 1o
            const float* s1p = &h1[(snd * MUL + m) * 3];
            const float b2 = wv * ey0;
            atomicAdd(&a1[(rcv * 128 + 64 + m) * 3 + 0], b2 * s1p[0]);
            atomicAdd(&a1[(rcv * 128 + 64 + m) * 3 + 1], b2 * s1p[1]);
            atomicAdd(&a1[(rcv * 128 + 64 + m) * 3 + 2], b2 * s1p[2]);
          }
        }
      }
      // DS ops are in-order per wave: the s_wj loads above complete before the
      // next j iteration's ds stores commit -> no extra sync needed here.
    }
  }
}

// ---------------------------------------------------------------------------
// Kernel 3: linear_down + species self-connection (all species, predicated
// add) + equivariant gate.  One wave per 16-node tile.
// ---------------------------------------------------------------------------
#define K3_WAVES 4
__global__ __launch_bounds__(K3_WAVES * 32) void k_down_gate(
    const float* __restrict__ a0, const float* __restrict__ a1,
    const float* __restrict__ x0, const float* __restrict__ x1,
    const int* __restrict__ species,
    const float* __restrict__ Wd0, const float* __restrict__ Wd1,
    const float* __restrict__ Wsc0, const float* __restrict__ Wsc1,
    float* __restrict__ out_sc, float* __restrict__ out_vec) {
  __shared__ float s_f0[K3_WAVES][16 * 128];  // 8 KB / wave
  __shared__ float s_f1[K3_WAVES][16 * 64];   // 4 KB / wave
  const int lane = threadIdx.x & 31;
  const int wave = threadIdx.x >> 5;
  const int half = lane >> 4;
  const int lcol = lane & 15;
  const int nwaves = K3_WAVES * gridDim.x;
  const int gwave = blockIdx.x * K3_WAVES + wave;
  const float inv2 = 0.08838834764831845f;  // 1/sqrt(2*MUL)
  const float inv = 0.125f;                  // 1/sqrt(MUL)

  for (int tile = gwave; tile < NN / 16; tile += nwaves) {
    const int n0 = tile * 16;
    // ---- d0 = (a0 @ Wd0)*inv2 into LDS ----
    for (int ct = 0; ct < 8; ++ct) {
      v8f acc = {};
      for (int k0 = 0; k0 < 128; k0 += 4) {
        const int kk = k0 + 2 * half;
        v2f a; a.x = a0[(n0 + lcol) * 128 + kk];
               a.y = a0[(n0 + lcol) * 128 + kk + 1];
        v2f b; b.x = Wd0[kk * 128 + ct * 16 + lcol];
               b.y = Wd0[(kk + 1) * 128 + ct * 16 + lcol];
        acc = wmma4(a, b, acc);
      }
#pragma unroll
      for (int i = 0; i < 8; ++i)
        s_f0[wave][(i + 8 * half) * 128 + ct * 16 + lcol] = acc[i] * inv2;
    }
    // ---- sc0: compute all 4 species, add rows whose species matches ----
    for (int s = 0; s < NSPEC; ++s) {
      for (int ct = 0; ct < 8; ++ct) {
        v8f acc = {};
        for (int k0 = 0; k0 < MUL; k0 += 4) {
          const int kk = k0 + 2 * half;
          v2f a; a.x = x0[(n0 + lcol) * MUL + kk];
                 a.y = x0[(n0 + lcol) * MUL + kk + 1];
          v2f b; b.x = Wsc0[(s * MUL + kk) * 128 + ct * 16 + lcol];
                 b.y = Wsc0[(s * MUL + kk + 1) * 128 + ct * 16 + lcol];
          acc = wmma4(a, b, acc);
        }
#pragma unroll
        for (int i = 0; i < 8; ++i) {
          const int row = i + 8 * half;
          if (species[n0 + row] == s)
            s_f0[wave][row * 128 + ct * 16 + lcol] += acc[i] * inv;
        }
      }
    }
    wave_lds_sync();
    // ---- scalars out; gates stored in-place in s_f0 upper half ----
    for (int idx = lane; idx < 16 * MUL; idx += 32) {
      const int row = idx >> 6, m = idx & 63;
      out_sc[(n0 + row) * MUL + m] = silu(0.5f * s_f0[wave][row * 128 + m]);
      s_f0[wave][row * 128 + 64 + m] =
          silu(0.5f * s_f0[wave][row * 128 + 64 + m]);
    }
    wave_lds_sync();
    // ---- per vector component: d1 + sc1, then gate-multiply ----
    for (int c = 0; c < 3; ++c) {
      for (int ct = 0; ct < 4; ++ct) {
        v8f acc = {};
        for (int k0 = 0; k0 < 128; k0 += 4) {
          const int kk = k0 + 2 * half;
          v2f a; a.x = a1[((n0 + lcol) * 128 + kk) * 3 + c];
                 a.y = a1[((n0 + lcol) * 128 + kk + 1) * 3 + c];
          v2f b; b.x = Wd1[kk * MUL + ct * 16 + lcol];
                 b.y = Wd1[(kk + 1) * MUL + ct * 16 + lcol];
          acc = wmma4(a, b, acc);
        }
#pragma unroll
        for (int i = 0; i < 8; ++i)
          s_f1[wave][(i + 8 * half) * MUL + ct * 16 + lcol] = acc[i] * inv2;
      }
      for (int s = 0; s < NSPEC; ++s) {
        for (int ct = 0; ct < 4; ++ct) {
          v8f acc = {};
          for (int k0 = 0; k0 < MUL; k0 += 4) {
            const int kk = k0 + 2 * half;
            v2f a; a.x = x1[((n0 + lcol) * MUL + kk) * 3 + c];
                   a.y = x1[((n0 + lcol) * MUL + kk + 1) * 3 + c];
            v2f b; b.x = Wsc1[(s * MUL + kk) * MUL + ct * 16 + lcol];
                   b.y = Wsc1[(s * MUL + kk + 1) * MUL + ct * 16 + lcol];
            acc = wmma4(a, b, acc);
          }
#pragma unroll
          for (int i = 0; i < 8; ++i) {
            const int row = i + 8 * half;
            if (species[n0 + row] == s)
              s_f1[wave][row * MUL + ct * 16 + lcol] += acc[i] * inv;
          }
        }
      }
      wave_lds_sync();
      for (int idx = lane; idx < 16 * MUL; idx += 32) {
        const int row = idx >> 6, m = idx & 63;
        const float g = s_f0[wave][row * 128 + 64 + m];
        out_vec[((n0 + row) * MUL + m) * 3 + c] =
            0.5f * s_f1[wave][row * MUL + m] * g;
      }
      wave_lds_sync();
    }
  }
}

// ---------------------------------------------------------------------------
extern "C" void kernel_launch(void* const* d_in, const int* in_sizes, int n_in,
                              void* d_out, int out_size, void* d_ws,
                              size_t ws_size, hipStream_t stream) {
  const float* x0 = (const float*)d_in[0];    // [N,64]
  const float* x1 = (const float*)d_in[1];    // [N,64,3]
  const float* ey0 = (const float*)d_in[2];   // [E]
  const float* ey1 = (const float*)d_in[3];   // [E,3]
  const float* re = (const float*)d_in[4];    // [E,8]
  const int* senders = (const int*)d_in[5];
  const int* receivers = (const int*)d_in[6];
  const int* species = (const int*)d_in[7];
  const float* Wup0 = (const float*)d_in[8];   // [64,64]
  const float* Wup1 = (const float*)d_in[9];   // [64,64]
  const float* Rw1 = (const float*)d_in[10];   // [8,64]
  const float* Rw2 = (const float*)d_in[11];   // [64,256]
  const float* Wd0 = (const float*)d_in[12];   // [128,128]
  const float* Wd1 = (const float*)d_in[13];   // [128,64]
  const float* Wsc0 = (const float*)d_in[14];  // [4,64,128]
  const float* Wsc1 = (const float*)d_in[15];  // [4,64,64]

  float* out = (float*)d_out;  // scalars [N,64] then vectors [N,64,3]

  // Workspace layout: h0 | h1 | a0 | a1  = N*(64+192+128+384) floats = 48 MB
  float* h0 = (float*)d_ws;
  float* h1 = h0 + (size_t)NN * 64;
  float* a0 = h1 + (size_t)NN * 192;
  float* a1 = a0 + (size_t)NN * 128;

  // zero the scatter accumulators each call (graph-capture legal)
  hipMemsetAsync(a0, 0, (size_t)NN * (128 + 384) * sizeof(float), stream);

  // 1024 node tiles -> 1024 waves (128 blocks x 8 waves)
  k_linear_up<<<128, 256, 0, stream>>>(x0, x1, Wup0, Wup1, h0, h1);
  // 16384 edge subtiles -> 4096 waves, 4 subtiles each (exact division)
  k_edge<<<512, K2_WAVES * 32, 0, stream>>>(re, Rw1, Rw2, h0, h1, ey0, ey1,
                                            senders, receivers, a0, a1);
  // 1024 node tiles -> 1024 waves (256 blocks x 4 waves)
  k_down_gate<<<256, K3_WAVES * 32, 0, stream>>>(
      a0, a1, x0, x1, species, Wd0, Wd1, Wsc0, Wsc1, out, out + (size_t)NN * 64);
}